// StochasticKeyNet_24489903522405
// MI455X (gfx1250) — compile-verified
//
#include <hip/hip_runtime.h>

typedef __attribute__((ext_vector_type(2))) float v2f;
typedef __attribute__((ext_vector_type(8))) float v8f;

#define BATCH 256

// ---------------------------------------------------------------------------
// u0[j,b] = x0[ip0[j], b] / s0[ip0[j]]      (the only *real* unkey)
// ---------------------------------------------------------------------------
__global__ __launch_bounds__(256)
void unkey0_kernel(const float* __restrict__ x0,
                   const int* __restrict__ ip0,
                   const float* __restrict__ s0,
                   float* __restrict__ u0)
{
    const int j = blockIdx.x;          // row (uniform per block -> scalar loads)
    const int b = threadIdx.x;         // batch column
    const int p = ip0[j];
    u0[(long)j * BATCH + b] = x0[(long)p * BATCH + b] / s0[p];
}

// ---------------------------------------------------------------------------
// Implicit-GEMM conv using V_WMMA_F32_16X16X4_F32.
// fp32 end-to-end: the net moves ~750MB for 144 GFLOP -> HBM-bound at
// 23.3 TB/s, so even the fp32 WMMA flavor is free; keeps reference precision.
//
// K reduction is reordered as (ky,kx) outer / ci inner (WMMA only needs A and
// B to agree on K-slot ordering) so the im2col decode is division-free and
// the halo test hoists out of the inner loop.
//
// One wave computes 16(cout) x 64(batch) at one (oy,ox): 4 accumulators share
// each A (weight) tile; B loads are coalesced (lanes 0-15 consecutive).
// Fused epilogue: + bias*hrow, (v*sa)/sa round-trip, ReLU, (v*sb)/sb
// round-trip, store u_next.  relu_mode==0 -> raw store (final FC2 layer).
// FC layers are 1x1 convs with h=ho=1.
// ---------------------------------------------------------------------------
__global__ __launch_bounds__(32)
void conv_wmma_kernel(const float* __restrict__ uin,
                      const float* __restrict__ W,
                      const float* __restrict__ bias,
                      const int*   __restrict__ ipA, const float* __restrict__ sA,
                      const int*   __restrict__ ipB, const float* __restrict__ sB,
                      float* __restrict__ uout,
                      int cin, int cout, int ksz, int st, int pad,
                      int h, int ho, int din, int relu_mode)
{
    const int lane = threadIdx.x;      // wave32
    const int half = lane >> 4;        // A/B K-slot pair: {0,1} or {2,3}
    const int m    = lane & 15;

    // block -> (coTile, oy, ox, bTile)
    int idx = blockIdx.x;
    const int nBT = BATCH / 64;        // 4 batch tiles of 64 columns
    const int bt  = idx % nBT; idx /= nBT;
    const int ox  = idx % ho;  idx /= ho;     // square maps: wo == ho, w == h
    const int oy  = idx % ho;  idx /= ho;
    const int coBase = idx * 16;
    const int bBase  = bt * 64;

    const int  K   = cin * ksz * ksz;
    const int  kk2 = ksz * ksz;
    const long hwB = (long)h * h * BATCH;     // row stride per input channel

    const int  co   = coBase + m;
    const bool coOK = (co < cout);
    const float* Wco = W + (long)co * K;      // dereferenced only when coOK

    v8f c0 = {}, c1 = {}, c2 = {}, c3 = {};   // 4 x 16 batch columns

    for (int ky = 0; ky < ksz; ++ky) {
        const int iy = oy * st + ky - pad;
        for (int kx = 0; kx < ksz; ++kx) {
            const int  ix  = ox * st + kx - pad;
            const bool ok  = (iy >= 0) & (iy < h) & (ix >= 0) & (ix < h);
            const int  woff = ky * ksz + kx;
            const float* bptr = uin + ((long)(iy * h + ix) * BATCH + bBase + m);

            for (int ci = 0; ci < cin; ci += 4) {
                const int cA = ci + 2 * half;   // this lane's two K slots

                // ---- A tile: 16(cout) x 4(ci) weights, zero-padded ----
                float a0 = 0.f, a1 = 0.f;
                if (coOK) {
                    if (cA     < cin) a0 = Wco[(long)cA * kk2 + woff];
                    if (cA + 1 < cin) a1 = Wco[(long)(cA + 1) * kk2 + woff];
                }
                v2f A; A.x = a0; A.y = a1;

                // ---- B tiles: 4(ci) x 64(batch), implicit im2col ----
                float b00 = 0.f, b01 = 0.f, b02 = 0.f, b03 = 0.f;
                float b10 = 0.f, b11 = 0.f, b12 = 0.f, b13 = 0.f;
                if (ok) {
                    if (cA < cin) {
                        const float* p = bptr + (long)cA * hwB;
                        b00 = p[0]; b01 = p[16]; b02 = p[32]; b03 = p[48];
                        if (cA + 4 < cin)                     // hide L2 latency
                            __builtin_prefetch(p + 4 * hwB, 0, 3);
                    }
                    if (cA + 1 < cin) {
                        const float* p = bptr + (long)(cA + 1) * hwB;
                        b10 = p[0]; b11 = p[16]; b12 = p[32]; b13 = p[48];
                    }
                }
                v2f B0; B0.x = b00; B0.y = b10;
                v2f B1; B1.x = b01; B1.y = b11;
                v2f B2; B2.x = b02; B2.y = b12;
                v2f B3; B3.x = b03; B3.y = b13;

                // D = A x B + C (fp32 16x16x4); EXEC is all-ones here.
                c0 = __builtin_amdgcn_wmma_f32_16x16x4_f32(false, A, false, B0,
                                                           (short)0, c0, false, false);
                c1 = __builtin_amdgcn_wmma_f32_16x16x4_f32(false, A, false, B1,
                                                           (short)0, c1, false, false);
                c2 = __builtin_amdgcn_wmma_f32_16x16x4_f32(false, A, false, B2,
                                                           (short)0, c2, false, false);
                c3 = __builtin_amdgcn_wmma_f32_16x16x4_f32(false, A, false, B3,
                                                           (short)0, c3, false, false);
            }
        }
    }

    // ---- fused epilogue ----
    const long ho_off = (long)(din - 1) * BATCH + bBase + m;
    const float hr0 = uin[ho_off];
    const float hr1 = uin[ho_off + 16];
    const float hr2 = uin[ho_off + 32];
    const float hr3 = uin[ho_off + 48];
    const int how = ho * ho;
#pragma unroll
    for (int r = 0; r < 8; ++r) {
        const int c = coBase + r + half * 8;    // C/D layout: VGPR r, M = r + 8*half
        if (c < cout) {
            const int j = c * how + oy * ho + ox;
            const float bv = bias[c];
            float v0 = c0[r] + bv * hr0;
            float v1 = c1[r] + bv * hr1;
            float v2 = c2[r] + bv * hr2;
            float v3 = c3[r] + bv * hr3;
            if (relu_mode) {
                const float sa = sA[ipA[j]];          // key_a -> unkey_a round-trip
                v0 = (v0 * sa) / sa;  v1 = (v1 * sa) / sa;
                v2 = (v2 * sa) / sa;  v3 = (v3 * sa) / sa;
                v0 = fmaxf(v0, 0.f);  v1 = fmaxf(v1, 0.f);
                v2 = fmaxf(v2, 0.f);  v3 = fmaxf(v3, 0.f);
                const float sb = sB[ipB[j]];          // key_b -> unkey_b round-trip
                v0 = (v0 * sb) / sb;  v1 = (v1 * sb) / sb;
                v2 = (v2 * sb) / sb;  v3 = (v3 * sb) / sb;
            }
            float* o = uout + (long)j * BATCH + bBase + m;
            o[0]  = v0;
            o[16] = v1;
            o[32] = v2;
            o[48] = v3;
        }
    }
}

// ---------------------------------------------------------------------------
// Homogeneous row: u_next[dout-1, b] = roundtrip_b(relu(roundtrip_a(u[din-1,b])))
// ---------------------------------------------------------------------------
__global__ __launch_bounds__(256)
void hrow_kernel(const float* __restrict__ uin, int din,
                 float* __restrict__ uout, int dout,
                 const int* __restrict__ ipA, const float* __restrict__ sA,
                 const int* __restrict__ ipB, const float* __restrict__ sB,
                 int relu_mode)
{
    const int b = threadIdx.x;
    float v = uin[(long)(din - 1) * BATCH + b];
    const int j = dout - 1;
    if (relu_mode) {
        const float sa = sA[ipA[j]];
        v = (v * sa) / sa;
        v = fmaxf(v, 0.f);
        const float sb = sB[ipB[j]];
        v = (v * sb) / sb;
    }
    uout[(long)j * BATCH + b] = v;
}

// ---------------------------------------------------------------------------
// Final real key: out[i,b] = s11[i] * z[p11[i], b]   (i in 0..10)
// ---------------------------------------------------------------------------
__global__ __launch_bounds__(256)
void final_key_kernel(const float* __restrict__ z,
                      const int* __restrict__ p11,
                      const float* __restrict__ s11,
                      float* __restrict__ out)
{
    const int i = blockIdx.x;
    const int b = threadIdx.x;
    out[i * BATCH + b] = s11[i] * z[(long)p11[i] * BATCH + b];
}

// ---------------------------------------------------------------------------
// Input index map (setup_inputs dict order, nested dicts in insertion order):
//  0        : A0x0_affine (3073 x 256 f32)
//  1..9     : conv_w conv1..conv9
//  10..18   : conv_b conv1..conv9
//  19,20    : fc1_w (100x640), fc1_b
//  21,22    : fc2_w (10x100),  fc2_b
//  23..44   : perms   [A0, A11, A1a, A1b, ..., A10a, A10b]
//  45..66   : iperms  [same order]
//  67..88   : scales  [same order]
// ---------------------------------------------------------------------------
extern "C" void kernel_launch(void* const* d_in, const int* in_sizes, int n_in,
                              void* d_out, int out_size, void* d_ws, size_t ws_size,
                              hipStream_t stream)
{
    (void)in_sizes; (void)n_in; (void)out_size; (void)ws_size;

    const float* x0 = (const float*)d_in[0];
    const float* Wc[9]; const float* Bc[9];
    for (int i = 0; i < 9; ++i) { Wc[i] = (const float*)d_in[1 + i]; Bc[i] = (const float*)d_in[10 + i]; }
    const float* fc1w = (const float*)d_in[19];
    const float* fc1b = (const float*)d_in[20];
    const float* fc2w = (const float*)d_in[21];
    const float* fc2b = (const float*)d_in[22];

    const int*   p11 = (const int*)d_in[24];     // perms['A11']
    const int*   ip0 = (const int*)d_in[45];     // iperms['A0']
    const float* s0  = (const float*)d_in[67];   // scales['A0']
    const float* s11 = (const float*)d_in[68];   // scales['A11']

    // per-layer key arrays (reference layer number i = 1..10)
    auto ipa = [&](int i) { return (const int*)d_in[47 + 2 * (i - 1)]; };   // iperms[Aia]
    auto ipb = [&](int i) { return (const int*)d_in[48 + 2 * (i - 1)]; };   // iperms[Aib]
    auto sca = [&](int i) { return (const float*)d_in[69 + 2 * (i - 1)]; }; // scales[Aia]
    auto scb = [&](int i) { return (const float*)d_in[70 + 2 * (i - 1)]; }; // scales[Aib]

    // workspace: two ping-pong activation slots (max tensor 98305 x 256 f32) + z
    const long CAP = 98305L * BATCH;
    float* slot[2] = { (float*)d_ws, (float*)d_ws + CAP };
    float* zfin    = (float*)d_ws + 2 * CAP;     // 11 x 256

    struct L { int cin, cout, k, st, pad, h, ho, din, dout; const float *W, *B; };
    const L Ls[11] = {
        {  3,  96, 3, 1, 1, 32, 32,  3073, 98305, Wc[0], Bc[0] },
        { 96,  96, 3, 1, 1, 32, 32, 98305, 98305, Wc[1], Bc[1] },
        { 96,  96, 3, 2, 1, 32, 16, 98305, 24577, Wc[2], Bc[2] },
        { 96, 192, 3, 1, 1, 16, 16, 24577, 49153, Wc[3], Bc[3] },
        {192, 192, 3, 1, 1, 16, 16, 49153, 49153, Wc[4], Bc[4] },
        {192, 192, 3, 2, 1, 16,  8, 49153, 12289, Wc[5], Bc[5] },
        {192, 192, 3, 1, 1,  8,  8, 12289, 12289, Wc[6], Bc[6] },
        {192, 192, 1, 1, 0,  8,  8, 12289, 12289, Wc[7], Bc[7] },
        {192,  10, 1, 1, 0,  8,  8, 12289,   641, Wc[8], Bc[8] },
        {640, 100, 1, 1, 0,  1,  1,   641,   101, fc1w,  fc1b  },  // fc1 as 1x1 conv
        {100,  10, 1, 1, 0,  1,  1,   101,    11, fc2w,  fc2b  },  // fc2, raw epilogue
    };

    unkey0_kernel<<<3073, 256, 0, stream>>>(x0, ip0, s0, slot[0]);

    int cur = 0;
    for (int li = 0; li < 11; ++li) {
        const L& l = Ls[li];
        float* uin  = slot[cur];
        float* uout = (li == 10) ? zfin : slot[cur ^ 1];
        const int relu_mode = (li < 10) ? 1 : 0;
        const int ref = li + 1;                  // reference layer number
        const int*   pA = relu_mode ? ipa(ref) : nullptr;
        const int*   pB = relu_mode ? ipb(ref) : nullptr;
        const float* vA = relu_mode ? sca(ref) : nullptr;
        const float* vB = relu_mode ? scb(ref) : nullptr;

        const int coT  = (l.cout + 15) / 16;
        const int grid = coT * l.ho * l.ho * (BATCH / 64);
        conv_wmma_kernel<<<grid, 32, 0, stream>>>(uin, l.W, l.B, pA, vA, pB, vB,
                                                  uout, l.cin, l.cout, l.k, l.st,
                                                  l.pad, l.h, l.ho, l.din, relu_mode);
        hrow_kernel<<<1, 256, 0, stream>>>(uin, l.din, uout, l.dout,
                                           pA, vA, pB, vB, relu_mode);
        cur ^= 1;
    }

    final_key_kernel<<<11, 256, 0, stream>>>(zfin, p11, s11, (float*)d_out);
}